// KPNeXt_3822520893925
// MI455X (gfx1250) — compile-verified
//
#include <hip/hip_runtime.h>
#include <hip/hip_bf16.h>

// Problem constants (from reference)
#define NPTS 16384
#define HNBR 16
#define CCH  192
#define KKP  43
#define KKPP 44      // padded K for 4-wide WMMA k-steps
#define DDEP 2
#define BN_EPS 1e-5f

typedef float v2f  __attribute__((ext_vector_type(2)));
typedef float v8f  __attribute__((ext_vector_type(8)));
typedef unsigned int u32x4 __attribute__((ext_vector_type(4)));
typedef int   i32x4 __attribute__((ext_vector_type(4)));
typedef int   i32x8 __attribute__((ext_vector_type(8)));

// ---------------------------------------------------------------------------
// GEMM: Y[N,192] = X[N,192] @ W[192,192], fused per-channel sum / sumsq stats.
// grid = N/16 blocks of 128 threads (4 waves). K-outer loop with 3 column-tile
// accumulators per wave: each A fragment (contiguous pair in LDS -> one
// ds_load_b64) feeds 3 WMMAs.
// ---------------------------------------------------------------------------
__global__ __launch_bounds__(128) void kpnext_gemm_stats(
    const float* __restrict__ X, const float* __restrict__ W,
    float* __restrict__ Y, float* __restrict__ stats)
{
    __shared__ float Xs[16 * CCH];
    const int row0 = blockIdx.x * 16;
    const int tid  = threadIdx.x;
    const int wave = tid >> 5;
    const int lane = tid & 31;
    const int half = lane >> 4;     // 0: lanes 0-15, 1: lanes 16-31
    const int l16  = lane & 15;

    // stage 16x192 input tile: one fully contiguous 3072-float block
    {
        const float4* Xg  = (const float4*)(X + row0 * CCH);
        float4*       Xs4 = (float4*)Xs;
        for (int i = tid; i < (16 * CCH) / 4; i += 128)
            Xs4[i] = Xg[i];
    }
    __syncthreads();

    const int c0 = wave * 16;           // wave owns columns c0, c0+64, c0+128
    v8f acc[3] = {};
    for (int s = 0; s < CCH / 4; ++s) {
        const int kr = s * 4 + half * 2;          // first of 2 contiguous K rows
        const v2f a = *(const v2f*)&Xs[l16 * CCH + kr];   // ds_load_b64
        const float* Wr = &W[kr * CCH + l16];
        v2f b0, b1, b2;
        b0.x = Wr[c0];         b0.y = Wr[CCH + c0];
        b1.x = Wr[c0 + 64];    b1.y = Wr[CCH + c0 + 64];
        b2.x = Wr[c0 + 128];   b2.y = Wr[CCH + c0 + 128];
        acc[0] = __builtin_amdgcn_wmma_f32_16x16x4_f32(false, a, false, b0,
                                                       (short)0, acc[0], false, false);
        acc[1] = __builtin_amdgcn_wmma_f32_16x16x4_f32(false, a, false, b1,
                                                       (short)0, acc[1], false, false);
        acc[2] = __builtin_amdgcn_wmma_f32_16x16x4_f32(false, a, false, b2,
                                                       (short)0, acc[2], false, false);
    }

    #pragma unroll
    for (int j = 0; j < 3; ++j) {
        const int c = c0 + j * 64;
        float s1 = 0.f, s2 = 0.f;
        #pragma unroll
        for (int r = 0; r < 8; ++r) {
            // D layout: vgpr r, half h -> row M = r + 8*h, col = c + l16
            const float v = acc[j][r];
            Y[(row0 + r + half * 8) * CCH + c + l16] = v;
            s1 += v;
            s2 += v * v;
        }
        s1 += __shfl_xor(s1, 16, 32);
        s2 += __shfl_xor(s2, 16, 32);
        if (half == 0) {
            atomicAdd(&stats[c + l16], s1);
            atomicAdd(&stats[CCH + c + l16], s2);
        }
    }
}

// ---------------------------------------------------------------------------
// In-place BN (batch stats from sum/sumsq) + ReLU over [N, C], float4-wide.
// ---------------------------------------------------------------------------
__global__ __launch_bounds__(256) void kpnext_bn_relu(
    float* __restrict__ X, const float* __restrict__ stats,
    const float* __restrict__ g, const float* __restrict__ b)
{
    const int i = blockIdx.x * blockDim.x + threadIdx.x;   // float4 index
    if (i >= (NPTS * CCH) / 4) return;
    const int c = (i * 4) % CCH;                           // 192 % 4 == 0
    float4 x = ((const float4*)X)[i];
    float* xv = &x.x;
    #pragma unroll
    for (int j = 0; j < 4; ++j) {
        const float m = stats[c + j] * (1.f / NPTS);
        const float v = stats[CCH + c + j] * (1.f / NPTS) - m * m;
        xv[j] = fmaxf((xv[j] - m) * rsqrtf(v + BN_EPS) * g[c + j] + b[c + j], 0.f);
    }
    ((float4*)X)[i] = x;
}

// ---------------------------------------------------------------------------
// KPConvD: per point n (one wave each):
//   infl[h,k] = max(1 - sqrt(max(|nb|^2 + |kp|^2 - 2 nb.kp, 0)), 0)
//   A[h,c]    = sum_k infl[h,k] * Wk[k,c]        (WMMA, 16x44 @ 44x192)
//   h2[n,c]   = sum_h h1[ref[n,h], c] * A[h,c]
// Wk staged into LDS by the Tensor Data Mover (1-D D#, 33 KB broadcast);
// influences generated directly in A-fragment layout; LDS float atomics for
// per-block BN stats.
// ---------------------------------------------------------------------------
__global__ __launch_bounds__(256) void kpnext_kpagg(
    const float* __restrict__ coord, const int* __restrict__ refidx,
    const float* __restrict__ kp, const float* __restrict__ Wk,
    const float* __restrict__ H1, float* __restrict__ H2,
    float* __restrict__ stats)
{
    __shared__ float WkS[KKPP * CCH];   // padded 44 x 192
    __shared__ float kpS[KKPP * 4];     // x,y,z,|kp|^2 (pad rows -> infl 0)
    __shared__ float statS[2 * CCH];

    const int tid = threadIdx.x;

#if __has_builtin(__builtin_amdgcn_tensor_load_to_lds)
    if (tid < 32) {
        // Tensor DMA descriptor: 1-D copy of 43*192 = 8256 f32 (33024 B)
        const unsigned long long ga = (unsigned long long)(uintptr_t)Wk;
        const unsigned int lds = (unsigned int)(uintptr_t)&WkS[0]; // low 32b = LDS offset
        u32x4 g0; i32x8 g1; i32x4 g2 = {}; i32x4 g3 = {}; i32x8 g4 = {};
        g0[0] = 1u;                                   // count=1 (valid descriptor)
        g0[1] = lds;                                  // lds_addr (bytes)
        g0[2] = (unsigned int)ga;                     // global_addr[31:0]
        g0[3] = (unsigned int)((ga >> 32) & 0x1FFFFFFu) | (2u << 30); // addr[56:32] | type=2
        g1[0] = 2 << 16;                              // data_size = 4 bytes
        g1[1] = (int)((KKP * CCH & 0xFFFF) << 16);    // tensor_dim0[15:0]
        g1[2] = (int)(KKP * CCH >> 16);               // tensor_dim0[31:16], tensor_dim1=0
        g1[3] = (int)((KKP * CCH) << 16);             // tile_dim0 = 8256
        g1[4] = 0;                                    // tile_dim1/2 unused (1-D)
        g1[5] = KKP * CCH;                            // tensor_dim0_stride[31:0]
        g1[6] = 0;
        g1[7] = 0;
        __builtin_amdgcn_tensor_load_to_lds(g0, g1, g2, g3, g4, 0);
    }
#else
    {
        const float4* Wg  = (const float4*)Wk;
        float4*       Ws4 = (float4*)WkS;
        for (int i = tid; i < (KKP * CCH) / 4; i += 256)
            Ws4[i] = Wg[i];
    }
#endif
    // zero the padded 44th row
    for (int i = tid; i < CCH; i += 256)
        WkS[KKP * CCH + i] = 0.f;
    for (int i = tid; i < KKPP; i += 256) {
        if (i < KKP) {
            const float x = kp[i * 3], y = kp[i * 3 + 1], z = kp[i * 3 + 2];
            kpS[i * 4 + 0] = x; kpS[i * 4 + 1] = y; kpS[i * 4 + 2] = z;
            kpS[i * 4 + 3] = x * x + y * y + z * z;
        } else {
            kpS[i * 4 + 0] = 0.f; kpS[i * 4 + 1] = 0.f; kpS[i * 4 + 2] = 0.f;
            kpS[i * 4 + 3] = 1e30f;   // huge distance -> influence 0
        }
    }
    for (int i = tid; i < 2 * CCH; i += 256) statS[i] = 0.f;
#if __has_builtin(__builtin_amdgcn_tensor_load_to_lds)
    if (tid < 32) __builtin_amdgcn_s_wait_tensorcnt(0);
#endif
    __syncthreads();

    const int wave = tid >> 5, lane = tid & 31;
    const int n    = blockIdx.x * 8 + wave;
    const int half = lane >> 4, l16 = lane & 15;

    // relative neighbor coords for h = l16 (both halves replicate)
    const int   nbh = refidx[n * HNBR + l16];
    const float cx = coord[n * 3], cy = coord[n * 3 + 1], cz = coord[n * 3 + 2];
    const float nx = coord[nbh * 3]     - cx;
    const float ny = coord[nbh * 3 + 1] - cy;
    const float nz = coord[nbh * 3 + 2] - cz;
    const float nn = nx * nx + ny * ny + nz * nz;

    // A fragments (influence), computed once, reused for all 12 column tiles
    v2f afrag[KKPP / 4];
    #pragma unroll
    for (int s = 0; s < KKPP / 4; ++s) {
        const int k0 = s * 4 + half * 2;
        const int k1 = k0 + 1;
        float d2 = nn + kpS[k0 * 4 + 3]
                 - 2.f * (nx * kpS[k0 * 4] + ny * kpS[k0 * 4 + 1] + nz * kpS[k0 * 4 + 2]);
        afrag[s].x = fmaxf(1.f - sqrtf(fmaxf(d2, 0.f)), 0.f);
        d2 = nn + kpS[k1 * 4 + 3]
           - 2.f * (nx * kpS[k1 * 4] + ny * kpS[k1 * 4 + 1] + nz * kpS[k1 * 4 + 2]);
        afrag[s].y = fmaxf(1.f - sqrtf(fmaxf(d2, 0.f)), 0.f);
    }

    // gather row bases for the 8 rows (neighbors) this lane reduces
    int gbase[8];
    #pragma unroll
    for (int r = 0; r < 8; ++r)
        gbase[r] = refidx[n * HNBR + r + half * 8] * CCH;

    for (int t = 0; t < CCH / 16; ++t) {
        const int c0 = t * 16;
        v8f acc = {};
        #pragma unroll
        for (int s = 0; s < KKPP / 4; ++s) {
            const int k0 = s * 4 + half * 2;
            v2f b;
            b.x = WkS[k0 * CCH + c0 + l16];
            b.y = WkS[(k0 + 1) * CCH + c0 + l16];
            acc = __builtin_amdgcn_wmma_f32_16x16x4_f32(
                false, afrag[s], false, b, (short)0, acc, false, false);
        }
        // multiply by gathered h1 rows, reduce over neighbors
        float s = 0.f;
        #pragma unroll
        for (int r = 0; r < 8; ++r)
            s += acc[r] * H1[gbase[r] + c0 + l16];
        s += __shfl_xor(s, 16, 32);
        if (half == 0) {
            H2[n * CCH + c0 + l16] = s;
            atomicAdd(&statS[c0 + l16], s);          // ds_add_f32
            atomicAdd(&statS[CCH + c0 + l16], s * s);
        }
    }
    __syncthreads();
    for (int i = tid; i < 2 * CCH; i += 256)
        atomicAdd(&stats[i], statS[i]);
}

// ---------------------------------------------------------------------------
// out = relu(identity + BN(h3)), float4-wide.
// ---------------------------------------------------------------------------
__global__ __launch_bounds__(256) void kpnext_residual_bn(
    const float* __restrict__ H3, const float* __restrict__ stats,
    const float* __restrict__ g, const float* __restrict__ b,
    const float* __restrict__ Fin, float* __restrict__ Fout)
{
    const int i = blockIdx.x * blockDim.x + threadIdx.x;   // float4 index
    if (i >= (NPTS * CCH) / 4) return;
    const int c = (i * 4) % CCH;
    float4 x = ((const float4*)H3)[i];
    float4 f = ((const float4*)Fin)[i];
    float* xv = &x.x;
    const float* fv = &f.x;
    #pragma unroll
    for (int j = 0; j < 4; ++j) {
        const float m = stats[c + j] * (1.f / NPTS);
        const float v = stats[CCH + c + j] * (1.f / NPTS) - m * m;
        const float y = (xv[j] - m) * rsqrtf(v + BN_EPS) * g[c + j] + b[c + j];
        xv[j] = fmaxf(fv[j] + y, 0.f);
    }
    ((float4*)Fout)[i] = x;
}

extern "C" void kernel_launch(void* const* d_in, const int* in_sizes, int n_in,
                              void* d_out, int out_size, void* d_ws, size_t ws_size,
                              hipStream_t stream) {
    const float* coord = (const float*)d_in[0];
    const float* feat  = (const float*)d_in[1];
    const int*   ridx  = (const int*)d_in[2];
    const float* kp    = (const float*)d_in[3];
    const float* W1    = (const float*)d_in[4];
    const float* Wk    = (const float*)d_in[5];
    const float* W3    = (const float*)d_in[6];
    const float* g1 = (const float*)d_in[7],  *b1 = (const float*)d_in[8];
    const float* g2 = (const float*)d_in[9],  *b2 = (const float*)d_in[10];
    const float* g3 = (const float*)d_in[11], *b3 = (const float*)d_in[12];

    float* ws  = (float*)d_ws;
    float* h1  = ws;                       // N*C
    float* h2  = h1 + NPTS * CCH;          // N*C
    float* h3  = h1;                       // h1 free after kpagg consumes it
    float* st1 = h2 + NPTS * CCH;          // 2*C each
    float* st2 = st1 + 2 * CCH;
    float* st3 = st2 + 2 * CCH;
    float* out = (float*)d_out;

    const int elem4_blocks = ((NPTS * CCH) / 4 + 255) / 256;

    for (int i = 0; i < DDEP; ++i) {
        const float* src = (i == 0) ? feat : out;   // identity / layer input
        (void)hipMemsetAsync(st1, 0, 2 * CCH * sizeof(float), stream);
        (void)hipMemsetAsync(st2, 0, 2 * CCH * sizeof(float), stream);
        (void)hipMemsetAsync(st3, 0, 2 * CCH * sizeof(float), stream);

        kpnext_gemm_stats<<<NPTS / 16, 128, 0, stream>>>(
            src, W1 + i * CCH * CCH, h1, st1);
        kpnext_bn_relu<<<elem4_blocks, 256, 0, stream>>>(
            h1, st1, g1 + i * CCH, b1 + i * CCH);

        kpnext_kpagg<<<NPTS / 8, 256, 0, stream>>>(
            coord, ridx, kp, Wk + i * KKP * CCH, h1, h2, st2);
        kpnext_bn_relu<<<elem4_blocks, 256, 0, stream>>>(
            h2, st2, g2 + i * CCH, b2 + i * CCH);

        kpnext_gemm_stats<<<NPTS / 16, 128, 0, stream>>>(
            h2, W3 + i * CCH * CCH, h3, st3);
        kpnext_residual_bn<<<elem4_blocks, 256, 0, stream>>>(
            h3, st3, g3 + i * CCH, b3 + i * CCH, src, out);
    }
}